// QNN_39058432589864
// MI455X (gfx1250) — compile-verified
//
#include <hip/hip_runtime.h>
#include <cstring>
#include <cstddef>

#define DIM_U   (1u << 18)
#define NWIRES  18
#define SEQ_LEN_ 16384
#define N_DATA_ 4
#define OUT_    64
#define BATCH_  32
#define CHUNK   512
#define YPB     4096

typedef __attribute__((ext_vector_type(2))) float v2f;
typedef __attribute__((ext_vector_type(8))) float v8f;

struct RxPass {
  int piv0, piv1, piv2;     // sorted pivot bit positions of span{m0,m1,m2}
  unsigned m0, m1, m2;      // storage-space pair masks (= L^-1 of the 3 logical wires)
  int th0, th1, th2;        // indices into cos/sin tables
};

struct PermCols { unsigned c[NWIRES]; };   // c[w] = (P^4)^-1 e_w

__device__ __forceinline__ unsigned insz(unsigned v, int p) {
  unsigned lm = (1u << p) - 1u;
  return ((v & ~lm) << 1) | (v & lm);
}

// ---- prep: cos/sin(theta/2) tables (72 each) ----------------------------
__global__ void prep_kernel(const float* __restrict__ qw,
                            float* __restrict__ cw, float* __restrict__ sw) {
  int i = threadIdx.x;
  if (i < 72) {
    float t = 0.5f * qw[i];
    cw[i] = cosf(t);
    sw[i] = sinf(t);
  }
}

// ---- init: closed-form state after layer-0 RX on one-hot inputs ---------
// S[x][k] layout: 16 complex (128B) contiguous per storage index x.
__global__ __launch_bounds__(256) void init_kernel(float2* __restrict__ S,
                                                   const float* __restrict__ cw,
                                                   const float* __restrict__ sw) {
  unsigned tid = blockIdx.x * 256u + threadIdx.x;
  unsigned k = tid & 15u;
  unsigned y = tid >> 4;
  float cl[NWIRES], sl[NWIRES];
#pragma unroll
  for (int w = 0; w < NWIRES; ++w) { cl[w] = cw[w]; sl[w] = sw[w]; }
  // wire w <-> bit (17-w); address wires start at |0>, data wires at bits of k
  unsigned missel = (y & ~15u) | ((y ^ k) & 15u);   // 1 = mismatch on that wire
  int m = __popc(missel);
  float mag = 1.0f;
#pragma unroll
  for (int w = 0; w < NWIRES; ++w)
    mag *= ((missel >> (17 - w)) & 1u) ? sl[w] : cl[w];
  float re, im;                       // phase (-i)^m
  switch (m & 3) {
    case 0:  re = mag;  im = 0.0f; break;
    case 1:  re = 0.0f; im = -mag; break;
    case 2:  re = -mag; im = 0.0f; break;
    default: re = 0.0f; im = mag;  break;
  }
  S[(size_t)y * 16u + k] = make_float2(re, im);
}

// ---- fused 3-gate RX pass: radix-8 butterfly per thread ------------------
__global__ __launch_bounds__(256) void rx3_kernel(float2* __restrict__ S,
                                                  const float* __restrict__ cw,
                                                  const float* __restrict__ sw,
                                                  RxPass P) {
  unsigned tid = blockIdx.x * 256u + threadIdx.x;
  unsigned k = tid & 15u;
  unsigned r = tid >> 4;                      // 2^15 coset representatives
  unsigned x = insz(insz(insz(r, P.piv0), P.piv1), P.piv2);
  unsigned xe[8];
#pragma unroll
  for (int e = 0; e < 8; ++e)
    xe[e] = x ^ ((e & 1) ? P.m0 : 0u) ^ ((e & 2) ? P.m1 : 0u) ^ ((e & 4) ? P.m2 : 0u);
  float re[8], im[8];
#pragma unroll
  for (int e = 0; e < 8; ++e) {
    float2 v = S[(size_t)xe[e] * 16u + k];
    re[e] = v.x; im[e] = v.y;
  }
  float c, s;
  // RX butterfly: a' = c*a - i*s*b ; b' = -i*s*a + c*b  (symmetric in a,b)
#define BFLY(A, B)                                              \
  { float ar = re[A], ai = im[A], br = re[B], bi = im[B];       \
    re[A] = fmaf(c, ar,  s * bi); im[A] = fmaf(c, ai, -s * br); \
    re[B] = fmaf(c, br,  s * ai); im[B] = fmaf(c, bi, -s * ar); }
  c = cw[P.th0]; s = sw[P.th0];
  BFLY(0,1) BFLY(2,3) BFLY(4,5) BFLY(6,7)
  c = cw[P.th1]; s = sw[P.th1];
  BFLY(0,2) BFLY(1,3) BFLY(4,6) BFLY(5,7)
  c = cw[P.th2]; s = sw[P.th2];
  BFLY(0,4) BFLY(1,5) BFLY(2,6) BFLY(3,7)
#undef BFLY
#pragma unroll
  for (int e = 0; e < 8; ++e)
    S[(size_t)xe[e] * 16u + k] = make_float2(re[e], im[e]);
}

// ---- final permutation fold + |amp|^2 -> probs[y][16k] -------------------
__global__ __launch_bounds__(256) void permsq_kernel(const float2* __restrict__ S,
                                                     float* __restrict__ probs,
                                                     PermCols P) {
  unsigned tid = blockIdx.x * 256u + threadIdx.x;
  unsigned k = tid & 15u;
  unsigned y = tid >> 4;
  unsigned x = 0u;
#pragma unroll
  for (int w = 0; w < NWIRES; ++w)
    x ^= ((y >> (17 - w)) & 1u) ? P.c[w] : 0u;
  float2 v = S[(size_t)x * 16u + k];
  probs[(size_t)y * 16u + k] = fmaf(v.x, v.x, v.y * v.y);
}

// ---- GEMM: logits16 = probs(16 x DIM) @ W^T via V_WMMA_F32_16X16X4_F32 ---
// grid = 4 o-tiles * (DIM/YPB) y-blocks; 128 threads = 4 waves.
// W tile staged in LDS with stride-17 padding; deterministic partial sums.
__global__ __launch_bounds__(128) void gemm_kernel(const float* __restrict__ probs,
                                                   const float* __restrict__ W,
                                                   float* __restrict__ part) {
  __shared__ float bt[CHUNK * 17];
  const int ot = blockIdx.x & 3;
  const int yb = blockIdx.x >> 2;
  const int obase = ot * 16;
  const unsigned ybase = (unsigned)yb * YPB;
  const int tid = threadIdx.x;
  const int wave = tid >> 5;
  const int lane = tid & 31;
  const int klane = lane & 15;
  const int khalf = lane >> 4;
  v8f acc0 = {0.f,0.f,0.f,0.f,0.f,0.f,0.f,0.f};
  v8f acc1 = {0.f,0.f,0.f,0.f,0.f,0.f,0.f,0.f};
  for (int cc = 0; cc < YPB / CHUNK; ++cc) {
    const unsigned cbase = ybase + (unsigned)cc * CHUNK;
    for (int idx = tid; idx < 16 * CHUNK; idx += 128) {
      int o  = idx >> 9;            // idx / CHUNK
      int yy = idx & (CHUNK - 1);
      bt[yy * 17 + o] = W[(size_t)(obase + o) * DIM_U + cbase + yy];
      if (cc + 1 < YPB / CHUNK)     // prefetch next chunk of W into caches
        __builtin_prefetch(&W[(size_t)(obase + o) * DIM_U + cbase + CHUNK + yy], 0, 3);
    }
    __syncthreads();
    const int wbase = wave * 128;   // each wave owns 128 y's of the chunk
#pragma unroll
    for (int it = 0; it < 16; ++it) {
      {
        int ya = wbase + it * 8 + 2 * khalf;              // K = 2*half + v
        v2f a, b;
        a.x = probs[(size_t)(cbase + ya) * 16 + klane];   // A: M=klane (state k)
        a.y = probs[(size_t)(cbase + ya + 1) * 16 + klane];
        b.x = bt[ya * 17 + klane];                        // B: N=klane (out ch)
        b.y = bt[(ya + 1) * 17 + klane];
        acc0 = __builtin_amdgcn_wmma_f32_16x16x4_f32(false, a, false, b,
                                                     (short)0, acc0, false, false);
      }
      {
        int ya = wbase + it * 8 + 4 + 2 * khalf;
        v2f a, b;
        a.x = probs[(size_t)(cbase + ya) * 16 + klane];
        a.y = probs[(size_t)(cbase + ya + 1) * 16 + klane];
        b.x = bt[ya * 17 + klane];
        b.y = bt[(ya + 1) * 17 + klane];
        acc1 = __builtin_amdgcn_wmma_f32_16x16x4_f32(false, a, false, b,
                                                     (short)0, acc1, false, false);
      }
    }
    __syncthreads();
  }
  acc0 = acc0 + acc1;
  // D layout: VGPR i, lanes 0-15 -> M=i, lanes 16-31 -> M=i+8; N=lane%16
  size_t base = ((size_t)(ot * 64 + yb) * 4 + wave) * 256;
#pragma unroll
  for (int i = 0; i < 8; ++i) part[base + (size_t)i * 32 + lane] = acc0[i];
}

// ---- deterministic reduce + bias + sigmoid + batch scatter ---------------
__global__ void final_kernel(const int* __restrict__ xin,
                             const float* __restrict__ part,
                             const float* __restrict__ bias,
                             float* __restrict__ out) {
  int tid = blockIdx.x * blockDim.x + threadIdx.x;
  if (tid >= BATCH_ * OUT_) return;
  int b = tid >> 6;
  int o = tid & 63;
  const int* xb = xin + (size_t)b * SEQ_LEN_ * N_DATA_;   // x[b, 0, :]
  int k = (xb[0] << 3) | (xb[1] << 2) | (xb[2] << 1) | xb[3];
  int i    = k & 7;
  int lane = ((k >> 3) << 4) | (o & 15);
  int ot   = o >> 4;
  float s = 0.0f;
  for (int yb = 0; yb < 64; ++yb)
#pragma unroll
    for (int w = 0; w < 4; ++w)
      s += part[((size_t)(ot * 64 + yb) * 4 + w) * 256 + (size_t)i * 32 + lane];
  s += bias[o];
  out[tid] = 1.0f / (1.0f + expf(-s));
}

extern "C" void kernel_launch(void* const* d_in, const int* in_sizes, int n_in,
                              void* d_out, int out_size, void* d_ws, size_t ws_size,
                              hipStream_t stream) {
  const int*   x    = (const int*)d_in[0];
  const float* qw   = (const float*)d_in[1];
  const float* W    = (const float*)d_in[2];
  const float* bias = (const float*)d_in[3];
  float* out = (float*)d_out;
  char* ws = (char*)d_ws;

  float2* S    = (float2*)ws;                              // 32 MiB state
  float* probs = (float*)(ws + (size_t)32 * 1024 * 1024);  // 16 MiB
  float* part  = (float*)(ws + (size_t)48 * 1024 * 1024);  //  1 MiB partials
  float* cw    = (float*)(ws + (size_t)49 * 1024 * 1024);  // 72 floats
  float* sw    = cw + 72;

  prep_kernel<<<1, 128, 0, stream>>>(qw, cw, sw);
  init_kernel<<<(DIM_U * 16) / 256, 256, 0, stream>>>(S, cw, sw);

  // Track L^-1 columns: C[w] = L^-1 e_w, L = cumulative CNOT-ring permutation.
  unsigned C[NWIRES];
  for (int w = 0; w < NWIRES; ++w) C[w] = 1u << (17 - w);
  auto ring = [](unsigned* Cc) {
    unsigned nc[NWIRES];
    nc[0] = Cc[0] ^ Cc[1];
    for (int w = 1; w <= 16; ++w) nc[w] = Cc[w] ^ Cc[w + 1];
    nc[17] = Cc[0] ^ Cc[1] ^ Cc[17];
    memcpy(Cc, nc, sizeof(nc));
  };
  ring(C);  // layer-0 ring (layer-0 RX gates folded into init_kernel)

  for (int l = 1; l < 4; ++l) {
    for (int g = 0; g < 6; ++g) {
      unsigned m[3] = {C[3 * g], C[3 * g + 1], C[3 * g + 2]};
      unsigned t[3] = {m[0], m[1], m[2]};
      int piv[3] = {0, 0, 0};
      for (int i = 0; i < 3; ++i) {          // GF(2) elimination for pivots
        for (int j = 0; j < i; ++j) if ((t[i] >> piv[j]) & 1u) t[i] ^= t[j];
        piv[i] = 31 - __builtin_clz(t[i]);
        for (int j = 0; j < i; ++j) if ((t[j] >> piv[i]) & 1u) t[j] ^= t[i];
      }
      if (piv[0] > piv[1]) { int tt = piv[0]; piv[0] = piv[1]; piv[1] = tt; }
      if (piv[1] > piv[2]) { int tt = piv[1]; piv[1] = piv[2]; piv[2] = tt; }
      if (piv[0] > piv[1]) { int tt = piv[0]; piv[0] = piv[1]; piv[1] = tt; }
      RxPass P;
      P.piv0 = piv[0]; P.piv1 = piv[1]; P.piv2 = piv[2];
      P.m0 = m[0]; P.m1 = m[1]; P.m2 = m[2];
      P.th0 = l * 18 + 3 * g; P.th1 = P.th0 + 1; P.th2 = P.th0 + 2;
      rx3_kernel<<<2048, 256, 0, stream>>>(S, cw, sw, P);
    }
    ring(C);
  }

  PermCols PC;
  for (int w = 0; w < NWIRES; ++w) PC.c[w] = C[w];
  permsq_kernel<<<(DIM_U * 16) / 256, 256, 0, stream>>>(S, probs, PC);
  gemm_kernel<<<4 * (DIM_U / YPB), 128, 0, stream>>>(probs, W, part);
  final_kernel<<<(BATCH_ * OUT_ + 255) / 256, 256, 0, stream>>>(x, part, bias, out);
}